// OneHotBERT_20366734917666
// MI455X (gfx1250) — compile-verified
//
#include <hip/hip_runtime.h>

// ---------------------------------------------------------------------------
// OneHotBERT head on MI455X (gfx1250).
//   logits = ((E / rowsum(E)) * mask) @ W_embed^T @ W_cls^T + b
// Folded:  W_comb[2][V] = W_cls @ (W_embed * mask)   (tiny prep kernel)
//          logits[r][l] = (E[r] . W_comb[l]) / (E[r] . ones) + b[l]
// Main kernel: single HBM-bound sweep of the 1.65 GB fp32 embedding
// (roofline ~71us @ 23.3 TB/s) using V_WMMA_F32_16X16X4_F32 (exact fp32),
// unrolled 8x (one 128B line / lane / macro-iter) with dual accumulators
// and one branchless global_prefetch per macro-iter.
// ---------------------------------------------------------------------------

#define VOCAB   50265
#define HIDDEN  768
#define NLABELS 2
#define ROWS    (2 * 4096)              // B*S = 8192
#define VPAD    (((VOCAB + 3) / 4) * 4) // 50268
#define NCHUNK  (VPAD / 4)              // 12567 K-chunks of 4
#define FULLCH  (VOCAB / 4)             // 12566 full chunks (tail has 1 valid col)

typedef __attribute__((ext_vector_type(2))) float v2f;
typedef __attribute__((ext_vector_type(8))) float v8f;

// --------------------------------------------------------------------------
// Kernel 1: W_comb[l][v] = sum_h W_cls[l*H+h] * W_embed[h*V+v] * mask[v],
// zero-padded to VPAD. ~150 MB read, coalesced over v.
// --------------------------------------------------------------------------
__global__ __launch_bounds__(256)
void onehot_prep_wcomb(const float* __restrict__ W_embed,
                       const float* __restrict__ token_mask,
                       const float* __restrict__ W_cls,
                       float* __restrict__ Wc) {
    __shared__ float cls[NLABELS * HIDDEN];
    for (int i = threadIdx.x; i < NLABELS * HIDDEN; i += blockDim.x)
        cls[i] = W_cls[i];
    __syncthreads();

    const int v = blockIdx.x * blockDim.x + threadIdx.x;
    if (v >= VPAD) return;

    float c0 = 0.f, c1 = 0.f;
    if (v < VOCAB) {
        const float tm = token_mask[v];
        for (int h = 0; h < HIDDEN; ++h) {
            const float w = W_embed[(size_t)h * VOCAB + v];
            c0 = fmaf(cls[h], w, c0);
            c1 = fmaf(cls[HIDDEN + h], w, c1);
        }
        c0 *= tm; c1 *= tm;
    }
    Wc[v]        = c0;
    Wc[VPAD + v] = c1;
}

// --------------------------------------------------------------------------
// Kernel 2: one block per 16-row tile; 8 waves split the vocab (K) range.
// WMMA A = 16 rows x 4 K (fp32 embedding), B cols: {Wc0, Wc1, ones, 0...}.
// A layout (ISA 7.12.2): lane = M + 16*(K>=2), vgpr j = K%2.
// B layout (by symmetry): lane = N + 16*(K>=2), vgpr j = K%2.
// D layout: vgpr r, lane l -> row M = r + 8*(l>>4), col N = l&15.
// --------------------------------------------------------------------------
__global__ __launch_bounds__(256)
void onehot_reduce_wmma(const float* __restrict__ emb,
                        const float* __restrict__ Wc,
                        const float* __restrict__ b_cls,
                        float* __restrict__ out) {
    __shared__ float red[16 * 3];

    const int tid  = threadIdx.x;
    const int wave = tid >> 5;
    const int lane = tid & 31;
    const int n    = lane & 15;   // N index (B/D column) / A row within tile
    const int half = lane >> 4;
    const int kl   = half * 2;    // per-lane K sub-offset {0,2}

    if (tid < 48) red[tid] = 0.f;
    __syncthreads();

    const int    tile    = blockIdx.x;                       // 512 tiles
    const size_t rowbase = (size_t)(tile * 16 + n) * VOCAB;  // A row for this lane

    // blocked K-partition across the 8 waves (uniform within a wave)
    const int per = (NCHUNK + 7) / 8;        // 1571
    const int c0  = wave * per;
    const int c1  = (c0 + per < NCHUNK) ? (c0 + per) : NCHUNK;
    const int cf  = (c1 < FULLCH) ? c1 : FULLCH;   // full (unguarded) chunks

    const float* aptr = emb + rowbase + kl;                       // 4B aligned (V odd)
    const int    bn   = (n < 2) ? n : 0;
    const float* bptr = Wc + (size_t)bn * VPAD + kl;              // 8B aligned (VPAD%4==0)
    const bool   buse = (n < 2);
    const float  bcon = (n == 2) ? 1.0f : 0.0f;                   // ones column -> rowsum

    v8f acc0 = {};
    v8f acc1 = {};

    int c = c0;
    // main loop: 8 chunks = 32 floats = one 128B line per lane per macro-iter
    for (; c + 8 <= cf; c += 8) {
        __builtin_prefetch(aptr + c * 4 + 256, 0, 1);   // ~8 lines ahead, branchless
#pragma unroll
        for (int u = 0; u < 8; ++u) {
            const int k = (c + u) * 4;
            v2f a;
            a.x = aptr[k];
            a.y = aptr[k + 1];
            const v2f bl = *(const v2f*)(bptr + k);
            v2f b;
            b.x = buse ? bl.x : bcon;
            b.y = buse ? bl.y : bcon;
            if (u & 1)
                acc1 = __builtin_amdgcn_wmma_f32_16x16x4_f32(
                           false, a, false, b, (short)0, acc1, false, false);
            else
                acc0 = __builtin_amdgcn_wmma_f32_16x16x4_f32(
                           false, a, false, b, (short)0, acc0, false, false);
        }
    }
    // remainder full chunks
    for (; c < cf; ++c) {
        const int k = c * 4;
        v2f a;
        a.x = aptr[k];
        a.y = aptr[k + 1];
        const v2f bl = *(const v2f*)(bptr + k);
        v2f b;
        b.x = buse ? bl.x : bcon;
        b.y = buse ? bl.y : bcon;
        acc0 = __builtin_amdgcn_wmma_f32_16x16x4_f32(
                   false, a, false, b, (short)0, acc0, false, false);
    }

    // fold the two accumulators (exact fp32 adds)
#pragma unroll
    for (int r = 0; r < 8; ++r) acc0[r] += acc1[r];

    // tail chunk(s): element-guarded (selects only; EXEC stays all-ones for WMMA)
    const int ct = (c0 > FULLCH) ? c0 : FULLCH;
    for (int t = ct; t < c1; ++t) {
        const int k = t * 4;
        v2f a;
        a.x = (k + kl + 0 < VOCAB) ? aptr[k]     : 0.0f;
        a.y = (k + kl + 1 < VOCAB) ? aptr[k + 1] : 0.0f;
        const v2f bl = *(const v2f*)(bptr + k);   // Wc zero-padded, always safe
        const float g0 = (n == 2 && (k + kl + 0) < VOCAB) ? 1.0f : 0.0f;
        const float g1 = (n == 2 && (k + kl + 1) < VOCAB) ? 1.0f : 0.0f;
        v2f b;
        b.x = buse ? bl.x : g0;
        b.y = buse ? bl.y : g1;
        acc0 = __builtin_amdgcn_wmma_f32_16x16x4_f32(
                   false, a, false, b, (short)0, acc0, false, false);
    }

    // cross-wave reduction of the 16x3 useful D columns via LDS float atomics
#pragma unroll
    for (int r = 0; r < 8; ++r) {
        if (n < 3) {
            const int m = r + 8 * half;
            atomicAdd(&red[m * 3 + n], acc0[r]);
        }
    }
    __syncthreads();

    if (tid < 16) {
        const float d0  = red[tid * 3 + 0];
        const float d1  = red[tid * 3 + 1];
        const float inv = 1.0f / red[tid * 3 + 2];
        const int   row = tile * 16 + tid;
        out[row * 2 + 0] = fmaf(d0, inv, b_cls[0]);
        out[row * 2 + 1] = fmaf(d1, inv, b_cls[1]);
    }
}

// --------------------------------------------------------------------------
extern "C" void kernel_launch(void* const* d_in, const int* in_sizes, int n_in,
                              void* d_out, int out_size, void* d_ws, size_t ws_size,
                              hipStream_t stream) {
    (void)in_sizes; (void)n_in; (void)out_size; (void)ws_size;
    const float* emb        = (const float*)d_in[0];  // (B,S,V) fp32
    // d_in[1]: attention_masks (unused by the math)
    const float* W_embed    = (const float*)d_in[2];  // (H,V) fp32
    const float* token_mask = (const float*)d_in[3];  // (V,)  fp32
    const float* W_cls      = (const float*)d_in[4];  // (2,H) fp32
    const float* b_cls      = (const float*)d_in[5];  // (2,)  fp32
    float*       out        = (float*)d_out;          // (B,S,2) fp32
    float*       Wc         = (float*)d_ws;           // 2*VPAD floats (~402 KB)

    onehot_prep_wcomb<<<(VPAD + 255) / 256, 256, 0, stream>>>(
        W_embed, token_mask, W_cls, Wc);
    onehot_reduce_wmma<<<ROWS / 16, 256, 0, stream>>>(
        emb, Wc, b_cls, out);
}